// Criterion_5265629905388
// MI455X (gfx1250) — compile-verified
//
#include <hip/hip_runtime.h>

typedef float v2f __attribute__((ext_vector_type(2)));
typedef float v8f __attribute__((ext_vector_type(8)));

#define TRIP_MARGIN 1.0f

// ---------------------------------------------------------------------------
// Kernel 1: compute total = sum(img_num_per_cls), zero accumulator + output.
// Runs every launch, so graph replay re-initializes state.
// ---------------------------------------------------------------------------
__global__ void tl_init_kernel(const float* __restrict__ img_num, int C,
                               float* __restrict__ ws, float* __restrict__ out) {
    if (blockIdx.x == 0 && threadIdx.x == 0) {
        float t = 0.f;
        for (int i = 0; i < C; ++i) t += img_num[i];
        ws[0] = t;    // total
        ws[1] = 0.f;  // loss accumulator
        out[0] = 0.f;
    }
}

// ---------------------------------------------------------------------------
// Kernel 2: main triplet loss via WMMA.
// Each wave handles groups of 16 triplets. For a group:
//   u = p - n, v = p + n - 2a   (per row, 128-dim)
//   d_ap - d_an = u . v  =  diag( U(16x128) * V^T(128x16) )
// computed as 32 chained V_WMMA_F32_16X16X4_F32 (two accumulators for ILP).
//
// f32 16x16x4 operand layout (per ISA 7.12.2): lanes 0-15 hold rows 0-15 with
// K-pair {kb, kb+1}; lanes 16-31 hold rows 0-15 with K-pair {kb+2, kb+3}.
// A (=U, row-major 16x4) and B (=V^T, 4x16 col-per-lane) index identically,
// so each lane loads its row's float2 from a, p, n and derives both operands.
// ---------------------------------------------------------------------------
__global__ __launch_bounds__(256) void tl_wmma_kernel(
    const float* __restrict__ batch,
    const int*   __restrict__ triplets,
    const int*   __restrict__ labels,
    const float* __restrict__ img_num,
    float*       __restrict__ ws,
    int T, int D)
{
    const float total = ws[0];

    const int lane   = threadIdx.x & 31;
    const int wave   = (int)((blockIdx.x * blockDim.x + threadIdx.x) >> 5);
    const int nwaves = (int)((gridDim.x * blockDim.x) >> 5);
    const int ngroups = (T + 15) >> 4;

    const int m     = lane & 15;        // matrix row handled by this lane
    const int khalf = (lane >> 4) << 1; // 0 for lanes 0-15, 2 for lanes 16-31

    float acc = 0.f;

    for (int g = wave; g < ngroups; g += nwaves) {
        const int base = g << 4;
        int t = base + m;
        if (t >= T) t = T - 1;          // tail clamp (EXEC stays full for WMMA)

        const int ia = triplets[3 * t + 0];
        const int ip = triplets[3 * t + 1];
        const int in_ = triplets[3 * t + 2];
        const float* __restrict__ pa = batch + (long)ia * D;
        const float* __restrict__ pp = batch + (long)ip * D;
        const float* __restrict__ pn = batch + (long)in_ * D;

        v8f c0 = {0.f, 0.f, 0.f, 0.f, 0.f, 0.f, 0.f, 0.f};
        v8f c1 = {0.f, 0.f, 0.f, 0.f, 0.f, 0.f, 0.f, 0.f};

        #pragma unroll
        for (int kb = 0; kb < 128; kb += 8) {   // D == 128
            {
                const int k = kb + khalf;
                v2f a2 = *(const v2f*)(pa + k);
                v2f p2 = *(const v2f*)(pp + k);
                v2f n2 = *(const v2f*)(pn + k);
                v2f u = p2 - n2;
                v2f v = p2 + n2 - 2.0f * a2;
                c0 = __builtin_amdgcn_wmma_f32_16x16x4_f32(
                        false, u, false, v, (short)0, c0, false, false);
            }
            {
                const int k = kb + 4 + khalf;
                v2f a2 = *(const v2f*)(pa + k);
                v2f p2 = *(const v2f*)(pp + k);
                v2f n2 = *(const v2f*)(pn + k);
                v2f u = p2 - n2;
                v2f v = p2 + n2 - 2.0f * a2;
                c1 = __builtin_amdgcn_wmma_f32_16x16x4_f32(
                        false, u, false, v, (short)0, c1, false, false);
            }
        }

        v8f c = c0 + c1;

        // Diagonal (i,i): lanes 0-7 hold it in VGPR 'lane'; lanes 24-31 in
        // VGPR 'lane-24' (rows 8-15). Other lanes hold no diagonal element.
        const bool valid = (lane < 8) || (lane >= 24);
        const int  r     = (lane < 8) ? lane : (lane - 24);
        float d = 0.f;
        #pragma unroll
        for (int i = 0; i < 8; ++i) d = (r == i) ? c[i] : d;

        const int mrow = (lane < 8) ? lane : (lane - 16); // diag row index
        const int trip = base + mrow;
        if (valid && trip < T) {
            const int anchor = triplets[3 * trip + 0];
            const int cls    = labels[anchor];
            const float w    = total / img_num[cls];
            float loss = d + TRIP_MARGIN;       // relu(d_ap - d_an + margin)
            loss = loss > 0.f ? loss : 0.f;
            acc += loss * w;
        }
    }

    // wave32 reduction
    #pragma unroll
    for (int off = 16; off > 0; off >>= 1)
        acc += __shfl_xor(acc, off, 32);
    if (lane == 0)
        atomicAdd(&ws[1], acc);
}

// ---------------------------------------------------------------------------
// Kernel 3: finalize mean.
// ---------------------------------------------------------------------------
__global__ void tl_finalize_kernel(const float* __restrict__ ws,
                                   float* __restrict__ out, float invT) {
    if (blockIdx.x == 0 && threadIdx.x == 0)
        out[0] = ws[1] * invT;
}

// ---------------------------------------------------------------------------
// Launch: inputs are (batch[B*D] f32, triplets[T*3] i32, labels[B] i32,
// img_num_per_cls[C] f32); output is one f32 scalar.
// ---------------------------------------------------------------------------
extern "C" void kernel_launch(void* const* d_in, const int* in_sizes, int n_in,
                              void* d_out, int out_size, void* d_ws, size_t ws_size,
                              hipStream_t stream) {
    const float* batch    = (const float*)d_in[0];
    const int*   triplets = (const int*)  d_in[1];
    const int*   labels   = (const int*)  d_in[2];
    const float* img_num  = (const float*)d_in[3];
    float*       out      = (float*)d_out;
    float*       ws       = (float*)d_ws;

    const int B = in_sizes[2];            // 4096
    const int D = in_sizes[0] / B;        // 128
    const int T = in_sizes[1] / 3;        // 262144
    const int C = in_sizes[3];            // 100
    (void)n_in; (void)out_size; (void)ws_size;

    tl_init_kernel<<<1, 32, 0, stream>>>(img_num, C, ws, out);

    const int threads = 256;              // 8 waves per block (wave32)
    const int ngroups = (T + 15) >> 4;
    int blocks = (ngroups + 7) / 8;       // ~1 group per wave upper bound
    if (blocks > 1024) blocks = 1024;     // grid-stride covers the rest
    if (blocks < 1) blocks = 1;
    tl_wmma_kernel<<<blocks, threads, 0, stream>>>(batch, triplets, labels,
                                                   img_num, ws, T, D);

    tl_finalize_kernel<<<1, 32, 0, stream>>>(ws, out, 1.0f / (float)T);
}